// MultiScaleROIEmbedder_64845416235533
// MI455X (gfx1250) — compile-verified
//
#include <hip/hip_runtime.h>
#include <hip/hip_bf16.h>
#include <math.h>

// ---------------------------------------------------------------------------
// MultiScaleROIEmbedder, restructured for MI455X (gfx1250):
//   pooled_raw = roi_align_mean(raw feats)        (gather kernel, per level)
//   tok_pre    = pooled_raw @ Wcat^T  + analytic pooled pos-enc + bias  (WMMA)
//   out        = LN(tok_pre) @ W1^T -> GELU -> @ W2^T                  (WMMA)
// Exploits linearity: 1x1-conv commutes with ROI-align, so we never project
// (or even touch most of) the full 470MB feature pyramid.
// GEMMs use V_WMMA_F32_16X16X4_F32 (exact fp32), one wave per 16x64 tile,
// pair-packed B so every operand fetch is a single b64.
// ---------------------------------------------------------------------------

#define NTOK   800    // B*K = 8*100
#define KROIS  100
#define DM     256
#define CTOT   3584   // 512 + 1024 + 2048

typedef __attribute__((ext_vector_type(2))) float v2f;
typedef __attribute__((ext_vector_type(8))) float v8f;

__device__ __forceinline__ float gelu_exact(float x) {
    return 0.5f * x * (1.0f + erff(x * 0.7071067811865476f));
}

// ---------------------------------------------------------------------------
// Kernel 1: ROI-align-mean over RAW channels of one pyramid level.
// grid.x = 800 ROIs, block = 256 threads (each thread strides channels).
// Writes pooled[n, chOff .. chOff+C) = mean over 36 bilinear samples.
// ---------------------------------------------------------------------------
__global__ void roi_pool_kernel(const float* __restrict__ feat,
                                const float* __restrict__ boxes,
                                const int* __restrict__ piw,
                                const int* __restrict__ pih,
                                int C, int HW, int chOff,
                                float* __restrict__ pooled)
{
    const int n = blockIdx.x;
    const int b = n / KROIS;
    const float* bx = boxes + (size_t)n * 4;

    const float Hf = (float)HW;
    const float s  = fminf(Hf / ((float)piw[0] + 1e-6f),
                           Hf / ((float)pih[0] + 1e-6f));
    const float x1 = bx[0] * s - 0.5f;
    const float y1 = bx[1] * s - 0.5f;
    const float bw = (bx[2] - bx[0]) * s * (1.0f / 3.0f);
    const float bh = (bx[3] - bx[1]) * s * (1.0f / 3.0f);

    int   yl[6], yh[6], xl[6], xh[6];
    float fy[6], fx[6];
    bool  vy[6], vx[6];
#pragma unroll
    for (int i = 0; i < 6; ++i) {
        const float off = 0.25f + 0.5f * (float)i;
        float cy = y1 + bh * off;
        vy[i] = (cy >= -1.0f) && (cy <= Hf);
        cy = fmaxf(cy, 0.0f);
        yl[i] = min((int)cy, HW - 1);
        yh[i] = min(yl[i] + 1, HW - 1);
        fy[i] = cy - (float)yl[i];

        float cx = x1 + bw * off;
        vx[i] = (cx >= -1.0f) && (cx <= Hf);
        cx = fmaxf(cx, 0.0f);
        xl[i] = min((int)cx, HW - 1);
        xh[i] = min(xl[i] + 1, HW - 1);
        fx[i] = cx - (float)xl[i];
    }

    for (int c = threadIdx.x; c < C; c += 256) {
        const float* fb = feat + ((size_t)b * C + c) * (size_t)(HW * HW);
        float acc = 0.0f;
#pragma unroll
        for (int iy = 0; iy < 6; ++iy) {
            if (!vy[iy]) continue;
            const float* r0 = fb + yl[iy] * HW;
            const float* r1 = fb + yh[iy] * HW;
            const float hy = 1.0f - fy[iy], ly = fy[iy];
#pragma unroll
            for (int ix = 0; ix < 6; ++ix) {
                if (!vx[ix]) continue;
                const float hx = 1.0f - fx[ix], lx = fx[ix];
                acc += hy * hx * r0[xl[ix]] + hy * lx * r0[xh[ix]]
                     + ly * hx * r1[xl[ix]] + ly * lx * r1[xh[ix]];
            }
        }
        pooled[(size_t)n * CTOT + chOff + c] = acc * (1.0f / 36.0f);
    }
}

// ---------------------------------------------------------------------------
// Kernel 2: analytic pooled positional encoding + projection biases.
// Channels d<128 depend only on y, d>=128 only on x, so the 36-sample bilinear
// average collapses to 6 one-dimensional interpolations of sin/cos terms.
// grid.x = 800 ROIs, block = 256 (one thread per output channel).
// ---------------------------------------------------------------------------
__device__ __forceinline__ void pos_accum(float bx1, float by1, float bx2, float by2,
                                          int H, float s, int d,
                                          const float* __restrict__ bp, float& acc)
{
    const float Hf = (float)H;
    const float x1 = bx1 * s - 0.5f, y1 = by1 * s - 0.5f;
    const float bw = (bx2 - bx1) * s * (1.0f / 3.0f);
    const float bh = (by2 - by1) * s * (1.0f / 3.0f);

    const bool useY = (d < 128);
    const int  dd   = useY ? d : (d - 128);
    // dim_t[j] = 10000^((j>>1)/32), j = dd>>1  ->  1/dim_t = exp(-ln(1e4)/32 * (dd>>2))
    const float invt = expf(-0.28782313662425576f * (float)(dd >> 2));
    const float invL = 1.0f / (float)(H - 1);
    const bool  isCos = (dd & 1);

    int   Vy = 0, Vx = 0;
    float S = 0.0f;
#pragma unroll
    for (int i = 0; i < 6; ++i) {
        const float off = 0.25f + 0.5f * (float)i;

        float cy = y1 + bh * off;
        const bool vy = (cy >= -1.0f) && (cy <= Hf);
        cy = fmaxf(cy, 0.0f);
        int   yl = min((int)cy, H - 1);
        int   yh = min(yl + 1, H - 1);
        float fy = cy - (float)yl;
        if (vy) {
            ++Vy;
            if (useY) {
                const float pl = (float)yl * invL * invt;
                const float ph = (float)yh * invL * invt;
                const float gl = isCos ? cosf(pl) : sinf(pl);
                const float gh = isCos ? cosf(ph) : sinf(ph);
                S += (1.0f - fy) * gl + fy * gh;
            }
        }

        float cx = x1 + bw * off;
        const bool vx = (cx >= -1.0f) && (cx <= Hf);
        cx = fmaxf(cx, 0.0f);
        int   xlo = min((int)cx, H - 1);
        int   xhi = min(xlo + 1, H - 1);
        float fx  = cx - (float)xlo;
        if (vx) {
            ++Vx;
            if (!useY) {
                const float pl = (float)xlo * invL * invt;
                const float ph = (float)xhi * invL * invt;
                const float gl = isCos ? cosf(pl) : sinf(pl);
                const float gh = isCos ? cosf(ph) : sinf(ph);
                S += (1.0f - fx) * gl + fx * gh;
            }
        }
    }
    acc += bp[d] * (float)(Vy * Vx) * (1.0f / 36.0f);
    acc += S * (float)(useY ? Vx : Vy) * (1.0f / 36.0f);
}

__global__ void pos_bias_kernel(const float* __restrict__ boxes,
                                const float* __restrict__ b3,
                                const float* __restrict__ b4,
                                const float* __restrict__ b5,
                                const int* __restrict__ piw,
                                const int* __restrict__ pih,
                                float* __restrict__ tokpre)
{
    const int n = blockIdx.x;
    const int d = threadIdx.x;
    const float* bx = boxes + (size_t)n * 4;
    const float x1 = bx[0], y1 = bx[1], x2 = bx[2], y2 = bx[3];
    const float iw = (float)piw[0] + 1e-6f;
    const float ih = (float)pih[0] + 1e-6f;

    float acc = 0.0f;
    pos_accum(x1, y1, x2, y2, 128, fminf(128.0f / iw, 128.0f / ih), d, b3, acc);
    pos_accum(x1, y1, x2, y2,  64, fminf( 64.0f / iw,  64.0f / ih), d, b4, acc);
    pos_accum(x1, y1, x2, y2,  32, fminf( 32.0f / iw,  32.0f / ih), d, b5, acc);
    tokpre[(size_t)n * DM + d] = acc;
}

// ---------------------------------------------------------------------------
// Kernel 3: pack-transpose.  W (O x I) row-major -> Bp ((I/2) x O) of float2,
// where Bp[kp][n] = { W[n][2kp], W[n][2kp+1] }.  This matches the WMMA f32
// B-fragment K-split (lane halves take K{0,1}/{2,3}) so the GEMM fetches its
// whole B fragment with one aligned b64 load.  Writes are lane-contiguous.
// ---------------------------------------------------------------------------
__global__ void packT_kernel(const float* __restrict__ in, float2* __restrict__ out,
                             int O, int I, int ldo, int kpairOff)
{
    const int idx = blockIdx.x * 256 + threadIdx.x;
    const int half = I >> 1;
    if (idx >= O * half) return;
    const int kp = idx / O;          // pair index (outer)
    const int n  = idx - kp * O;     // output channel (inner -> coalesced store)
    float2 v;
    v.x = in[(size_t)n * I + 2 * kp];
    v.y = in[(size_t)n * I + 2 * kp + 1];
    out[(size_t)(kpairOff + kp) * ldo + n] = v;
}

// ---------------------------------------------------------------------------
// Kernel 4: fp32 WMMA GEMM.  D = act( A(MxK) * B(KxN) + Cin + bias ).
// One wave32 per 16x64 output tile (4 accumulators), K stepped by 4 via
// V_WMMA_F32_16X16X4_F32.  Per k-step: 1 A b64 (reused x4) + 4 B b64 + 4 wmma.
// A layout: lanes 0-15 hold K={0,1}, lanes 16-31 hold K={2,3} (ISA 16x4 f32).
// C/D layout: VGPR r -> rows r (lanes 0-15) / r+8 (lanes 16-31), col = lane%16.
// ---------------------------------------------------------------------------
__global__ __launch_bounds__(32)
void wmma_gemm_f32(const float* __restrict__ A, const float2* __restrict__ Bp,
                   const float* __restrict__ Cin, const float* __restrict__ bias,
                   float* __restrict__ Out, int N, int K, int act)
{
    const int m0    = blockIdx.x * 16;
    const int nBase = blockIdx.y * 64;
    const int lane  = threadIdx.x & 31;
    const int half  = lane >> 4;
    const int lr    = lane & 15;

    v8f acc[4];
#pragma unroll
    for (int t = 0; t < 4; ++t)
#pragma unroll
        for (int r = 0; r < 8; ++r)
            acc[t][r] = Cin ? Cin[(size_t)(m0 + r + half * 8) * N + nBase + 16 * t + lr]
                            : 0.0f;

    const float*  arow = A + (size_t)(m0 + lr) * K + 2 * half;
    const float2* bcol = Bp + nBase + lr + (size_t)half * N;   // pair row (k/2)+half
    for (int k = 0; k < K; k += 4) {
        v2f a = *(const v2f*)(arow + k);                       // b64, 8B aligned
        const float2* brow = bcol + (size_t)(k >> 1) * N;
#pragma unroll
        for (int t = 0; t < 4; ++t) {
            const float2 bb = brow[16 * t];                    // single b64
            v2f bv; bv.x = bb.x; bv.y = bb.y;
            acc[t] = __builtin_amdgcn_wmma_f32_16x16x4_f32(
                         false, a, false, bv, (short)0, acc[t], false, false);
        }
    }

#pragma unroll
    for (int t = 0; t < 4; ++t) {
        const int col = nBase + 16 * t + lr;
        const float bs = bias ? bias[col] : 0.0f;
#pragma unroll
        for (int r = 0; r < 8; ++r) {
            float v = acc[t][r] + bs;
            if (act) v = gelu_exact(v);
            Out[(size_t)(m0 + r + half * 8) * N + col] = v;
        }
    }
}

// ---------------------------------------------------------------------------
// Kernel 5: LayerNorm over 256 channels per token (one block per token).
// ---------------------------------------------------------------------------
__global__ void ln_kernel(const float* __restrict__ x,
                          const float* __restrict__ g,
                          const float* __restrict__ b,
                          float* __restrict__ y)
{
    __shared__ float red[DM];
    const int n = blockIdx.x;
    const int d = threadIdx.x;
    const float v = x[(size_t)n * DM + d];

    red[d] = v;
    __syncthreads();
    for (int s = 128; s > 0; s >>= 1) {
        if (d < s) red[d] += red[d + s];
        __syncthreads();
    }
    const float mean = red[0] * (1.0f / DM);
    __syncthreads();

    const float dv = v - mean;
    red[d] = dv * dv;
    __syncthreads();
    for (int s = 128; s > 0; s >>= 1) {
        if (d < s) red[d] += red[d + s];
        __syncthreads();
    }
    const float var = red[0] * (1.0f / DM);

    y[(size_t)n * DM + d] = dv * rsqrtf(var + 1e-5f) * g[d] + b[d];
}

// ---------------------------------------------------------------------------
// Host-side orchestration (graph-capture safe: launches only).
// ---------------------------------------------------------------------------
extern "C" void kernel_launch(void* const* d_in, const int* in_sizes, int n_in,
                              void* d_out, int out_size, void* d_ws, size_t ws_size,
                              hipStream_t stream)
{
    (void)in_sizes; (void)n_in; (void)out_size; (void)ws_size;

    const float* p3    = (const float*)d_in[0];
    const float* p4    = (const float*)d_in[1];
    const float* p5    = (const float*)d_in[2];
    const float* boxes = (const float*)d_in[3];
    const float* W3    = (const float*)d_in[4];
    const float* b3    = (const float*)d_in[5];
    const float* W4    = (const float*)d_in[6];
    const float* b4    = (const float*)d_in[7];
    const float* W5    = (const float*)d_in[8];
    const float* b5    = (const float*)d_in[9];
    const float* ln_g  = (const float*)d_in[10];
    const float* ln_b  = (const float*)d_in[11];
    const float* W1    = (const float*)d_in[12];
    const float* b1    = (const float*)d_in[13];
    const float* W2    = (const float*)d_in[14];
    const float* b2    = (const float*)d_in[15];
    const int*   piw   = (const int*)d_in[16];
    const int*   pih   = (const int*)d_in[17];
    float*       out   = (float*)d_out;

    // Workspace layout (floats): ~18.2 MB total.
    float* ws     = (float*)d_ws;
    float* pooled = ws;                                   // 800 x 3584
    float* tokpre = pooled + (size_t)NTOK * CTOT;         // 800 x 256
    float* Wt     = tokpre + (size_t)NTOK * DM;           // (3584/2) x 256 float2
    float* W1t    = Wt     + (size_t)CTOT * DM;           // (256/2)  x 256 float2
    float* W2t    = W1t    + (size_t)DM * DM;             // (256/2)  x 256 float2
    float* lnbuf  = W2t    + (size_t)DM * DM;             // 800 x 256
    float* hbuf   = lnbuf  + (size_t)NTOK * DM;           // 800 x 256

    float2* WtP  = (float2*)Wt;
    float2* W1tP = (float2*)W1t;
    float2* W2tP = (float2*)W2t;

    // --- pack-transposed weights (tiny) ---
    packT_kernel<<<(DM * ( 512/2) + 255) / 256, 256, 0, stream>>>(W3, WtP, DM,  512, DM,   0);
    packT_kernel<<<(DM * (1024/2) + 255) / 256, 256, 0, stream>>>(W4, WtP, DM, 1024, DM, 256);
    packT_kernel<<<(DM * (2048/2) + 255) / 256, 256, 0, stream>>>(W5, WtP, DM, 2048, DM, 768);
    packT_kernel<<<(DM * (  DM/2) + 255) / 256, 256, 0, stream>>>(W1, W1tP, DM, DM, DM, 0);
    packT_kernel<<<(DM * (  DM/2) + 255) / 256, 256, 0, stream>>>(W2, W2tP, DM, DM, DM, 0);

    // --- analytic pos-enc + biases -> tok_pre ---
    pos_bias_kernel<<<NTOK, DM, 0, stream>>>(boxes, b3, b4, b5, piw, pih, tokpre);

    // --- ROI gather-pool of raw channels (the bandwidth-bound part) ---
    roi_pool_kernel<<<NTOK, 256, 0, stream>>>(p3, boxes, piw, pih,  512, 128,    0, pooled);
    roi_pool_kernel<<<NTOK, 256, 0, stream>>>(p4, boxes, piw, pih, 1024,  64,  512, pooled);
    roi_pool_kernel<<<NTOK, 256, 0, stream>>>(p5, boxes, piw, pih, 2048,  32, 1536, pooled);

    // --- projection GEMM: tok_pre += pooled (800x3584) @ Wt (3584x256) ---
    dim3 gtiles(NTOK / 16, DM / 64);
    wmma_gemm_f32<<<gtiles, 32, 0, stream>>>(pooled, WtP, tokpre, nullptr, tokpre, DM, CTOT, 0);

    // --- MLP: LN -> (x@W1^T + b1) -> GELU -> (@W2^T + b2) ---
    ln_kernel<<<NTOK, DM, 0, stream>>>(tokpre, ln_g, ln_b, lnbuf);
    wmma_gemm_f32<<<gtiles, 32, 0, stream>>>(lnbuf, W1tP, nullptr, b1, hbuf, DM, DM, 1);
    wmma_gemm_f32<<<gtiles, 32, 0, stream>>>(hbuf,  W2tP, nullptr, b2, out,  DM, DM, 0);
}